// MechanicsModel_47605417509455
// MI455X (gfx1250) — compile-verified
//
#include <hip/hip_runtime.h>
#include <stdint.h>

#define NPART 8192
#define JSPLIT_MAX 16
#define TJ 256          // j-tile staged in LDS
#define IPB 128         // i's per block = 8 waves * 16 rows
#define EPS 1e-12f

typedef __attribute__((ext_vector_type(2))) float v2f;
typedef __attribute__((ext_vector_type(8))) float v8f;

// ---------------------------------------------------------------- phase 0
// encR[j] = active ? radius : -radius   (sign bit carries the active flag,
// radius in [0.5,1.5) so never zero) -> lets the j-tile copy be a raw copy.
__global__ void morse_encode_kernel(const float* __restrict__ radius,
                                    const int* __restrict__ parent,
                                    float* __restrict__ encR) {
    int j = blockIdx.x * 256 + threadIdx.x;
    float r = radius[j];
    encR[j] = (parent[j] >= 0) ? r : -r;
}

// ---------------------------------------------------------------- helpers
__device__ __forceinline__ float pair_coef(float px, float py, float pz, float ri,
                                           float x, float y, float z, float enc,
                                           float C1, float c2n, bool isSelf) {
    float dx = px - x, dy = py - y, dz = pz - z;
    float d2 = fmaf(dx, dx, fmaf(dy, dy, dz * dz));
    d2 = fmaxf(d2, EPS);
    float inv  = __builtin_amdgcn_rsqf(d2);   // v_rsq_f32
    float dist = d2 * inv;                    // sqrt(d2)
    float rj   = fabsf(enc);
    float arg  = dist - (ri + rj);
    float e    = __builtin_amdgcn_exp2f(c2n * arg);  // exp(-a*(dist-req))
    float coef = C1 * e * (e - 1.0f) * inv;          // fmag / dist
    bool ok = (enc > 0.0f) && !isSelf;
    return ok ? coef : 0.0f;
}

__device__ __forceinline__ void async_b128_to_lds(uint32_t lds_off, const void* g) {
    asm volatile("global_load_async_to_lds_b128 %0, %1, off"
                 :: "v"(lds_off), "v"(g) : "memory");
}

// ---------------------------------------------------------------- phase 1
// Each wave owns a 16-row i-tile. Accumulate C(16x16) += Coef(16x4) x B(4x16)
// via V_WMMA_F32_16X16X4_F32; B cols 0..2 = p_j, col 3 = 1.0 (rowsum -> S).
__global__ __launch_bounds__(256)
void morse_partial_kernel(const float* __restrict__ pos,
                          const float* __restrict__ radius,
                          const float* __restrict__ encR,
                          const float* __restrict__ pWellWidth,
                          const float* __restrict__ pWellDepth,
                          float* __restrict__ partial,   // [js][NPART][4]
                          int jslice) {
    __shared__ float sPos[TJ * 3];   // raw xyz copy of tile
    __shared__ float sEnc[TJ];       // signed radius copy of tile

    const int tid  = threadIdx.x;
    const int lane = tid & 31;
    const int wave = tid >> 5;
    const int n    = lane & 15;   // A-row (M) and also B/C column for this lane
    const int hi   = lane >> 4;   // half-wave select (K pairs / row offset)

    // lane-constant, branchless selection masks for building the B operand:
    // B[k][n] = p_j.x*m0 + p_j.y*m1 + p_j.z*m2 + one3
    const float m0   = (n == 0) ? 1.0f : 0.0f;
    const float m1   = (n == 1) ? 1.0f : 0.0f;
    const float m2   = (n == 2) ? 1.0f : 0.0f;
    const float one3 = (n == 3) ? 1.0f : 0.0f;

    const float aw  = pWellWidth[0];
    const float Dw  = pWellDepth[0];
    const float C1  = 2.0f * Dw * aw;
    const float c2n = -aw * 1.4426950408889634f;  // -a * log2(e)

    const int iTile = blockIdx.x * IPB + wave * 16;
    const int ig    = iTile + n;                  // global i of this lane's A-row
    const float pix = pos[ig * 3 + 0];
    const float piy = pos[ig * 3 + 1];
    const float piz = pos[ig * 3 + 2];
    const float ri  = radius[ig];

    const int jBase0 = blockIdx.y * jslice;

    v8f acc = {};

    for (int t0 = 0; t0 < jslice; t0 += TJ) {
        const int jT = jBase0 + t0;

        // ---- async stage of the j-tile into LDS (ASYNCcnt path) ----
        // pos tile: TJ*12 = 3072 B -> 192 x b128 ; enc tile: 1024 B -> 64 x b128
        if (tid < 192) {
            uint32_t loff = (uint32_t)(uintptr_t)(&sPos[0]) + (uint32_t)tid * 16u;
            const char* g = (const char*)(pos + (size_t)jT * 3) + (size_t)tid * 16;
            async_b128_to_lds(loff, g);
        } else {
            int t = tid - 192;
            uint32_t loff = (uint32_t)(uintptr_t)(&sEnc[0]) + (uint32_t)t * 16u;
            const char* g = (const char*)(encR + jT) + (size_t)t * 16;
            async_b128_to_lds(loff, g);
        }
        asm volatile("s_wait_asynccnt 0" ::: "memory");
        __syncthreads();

        // ---- compute: K=4 j's per WMMA; this lane covers k = 2*hi, 2*hi+1 ----
#pragma unroll 4
        for (int jl = 0; jl < TJ; jl += 4) {
            const int j0 = jl + hi * 2;
            const int j1 = j0 + 1;
            float x0 = sPos[j0 * 3 + 0], y0 = sPos[j0 * 3 + 1], z0 = sPos[j0 * 3 + 2];
            float x1 = sPos[j1 * 3 + 0], y1 = sPos[j1 * 3 + 1], z1 = sPos[j1 * 3 + 2];
            float e0 = sEnc[j0], e1 = sEnc[j1];
            const int jg0 = jT + j0, jg1 = jT + j1;

            v2f A, B;
            A.x = pair_coef(pix, piy, piz, ri, x0, y0, z0, e0, C1, c2n, jg0 == ig);
            A.y = pair_coef(pix, piy, piz, ri, x1, y1, z1, e1, C1, c2n, jg1 == ig);
            // branchless B build: cols 0..2 = p_j components, col 3 = 1.0
            B.x = fmaf(x0, m0, fmaf(y0, m1, fmaf(z0, m2, one3)));
            B.y = fmaf(x1, m0, fmaf(y1, m1, fmaf(z1, m2, one3)));

            acc = __builtin_amdgcn_wmma_f32_16x16x4_f32(
                false, A, false, B, (short)0, acc, false, false);
        }
        __syncthreads();
    }

    // ---- write partials: lane holds column n for rows {hi*8 .. hi*8+7} ----
    if (n < 4) {
#pragma unroll
        for (int v = 0; v < 8; ++v) {
            int i = iTile + v + hi * 8;
            partial[((size_t)blockIdx.y * NPART + i) * 4 + n] = acc[v];
        }
    }
}

// ---------------------------------------------------------------- phase 2
__global__ void morse_reduce_kernel(const float* __restrict__ pos,
                                    const int* __restrict__ parent,
                                    const float* __restrict__ partial,
                                    float* __restrict__ out, int js) {
    int i = blockIdx.x * 256 + threadIdx.x;
    float Fx = 0.f, Fy = 0.f, Fz = 0.f, S = 0.f;
    for (int s = 0; s < js; ++s) {
        const float* p4 = partial + ((size_t)s * NPART + i) * 4;
        Fx += p4[0]; Fy += p4[1]; Fz += p4[2]; S += p4[3];
    }
    float px = pos[i * 3 + 0], py = pos[i * 3 + 1], pz = pos[i * 3 + 2];
    bool act = parent[i] >= 0;
    float fx = S * px - Fx, fy = S * py - Fy, fz = S * pz - Fz;
    out[i * 3 + 0] = px + (act ? fx : 0.f);
    out[i * 3 + 1] = py + (act ? fy : 0.f);
    out[i * 3 + 2] = pz + (act ? fz : 0.f);
}

// ---------------------------------------------------------------- launch
extern "C" void kernel_launch(void* const* d_in, const int* in_sizes, int n_in,
                              void* d_out, int out_size, void* d_ws, size_t ws_size,
                              hipStream_t stream) {
    const float* pos    = (const float*)d_in[0];
    const float* radius = (const float*)d_in[1];
    const int*   parent = (const int*)d_in[2];
    const float* ww     = (const float*)d_in[3];
    const float* wd     = (const float*)d_in[4];
    float* out = (float*)d_out;

    // ws layout: partial [js][NPART][4] floats, then encR [NPART] floats
    int js = JSPLIT_MAX;
    while (js > 1 &&
           ((size_t)js * NPART * 4 + NPART) * sizeof(float) > ws_size)
        js >>= 1;
    float* partial = (float*)d_ws;
    float* encR    = partial + (size_t)js * NPART * 4;  // 16B-aligned offset

    morse_encode_kernel<<<NPART / 256, 256, 0, stream>>>(radius, parent, encR);

    dim3 g1(NPART / IPB, js);
    morse_partial_kernel<<<g1, 256, 0, stream>>>(pos, radius, encR, ww, wd,
                                                 partial, NPART / js);

    morse_reduce_kernel<<<NPART / 256, 256, 0, stream>>>(pos, parent, partial,
                                                         out, js);
}